// NoTradeRegionRNN_9242769622587
// MI455X (gfx1250) — compile-verified
//
#include <hip/hip_runtime.h>
#include <stdint.h>
#include <stddef.h>

#ifndef __has_builtin
#define __has_builtin(x) 0
#endif

#if __has_builtin(__builtin_amdgcn_global_load_async_to_lds_b32)
#define USE_ASYNC 1
#else
#define USE_ASYNC 0
#endif

#define STAGES 8  // async ring depth; wait immediate = 4*(STAGES-1) = 28

typedef __attribute__((address_space(1))) int gas_int;              // global
typedef __attribute__((address_space(3))) int las_int;              // LDS
typedef __attribute__((address_space(3))) volatile const float lds_vcf;

// ---------- CDNA5 wait helpers ----------
__device__ __forceinline__ void wait_async_oldest_stage() {
#if USE_ASYNC
#if __has_builtin(__builtin_amdgcn_s_wait_asynccnt)
  __builtin_amdgcn_s_wait_asynccnt(4 * (STAGES - 1));
#else
  asm volatile("s_wait_asynccnt 28" ::: "memory");
#endif
#endif
}

__device__ __forceinline__ void wait_ds_zero() {
#if __has_builtin(__builtin_amdgcn_s_wait_dscnt)
  __builtin_amdgcn_s_wait_dscnt(0);
#else
  asm volatile("s_wait_dscnt 0" ::: "memory");
#endif
}

// ---------- async global -> LDS (per-lane 4B) ----------
__device__ __forceinline__ void async_ld_b32(const float* g, float* l) {
#if USE_ASYNC
  __builtin_amdgcn_global_load_async_to_lds_b32(
      (gas_int*)(void*)const_cast<float*>(g),
      (las_int*)(void*)l,
      0, 0);
#else
  (void)g; (void)l;
#endif
}

// ---------- per-step recurrence constants ----------
struct Consts {
  float ac, bd;
  // bounds fully folded: m = relu(fma(a, slope, ncs)); bound = base - relu(A - m)
  float ncs_lbx, A_lbx, base_lbx;   // ncs = -c*slope
  float ncs_ubx, A_ubx, base_ubx;
  float ncs_lby, A_lby, base_lby;
  float ncs_uby, A_uby, base_uby;
  float Wi0, Wi1, Wh0, Wh1, W10, W11, W20, W21;
};

__device__ __forceinline__ void step(float& hx, float& hy,
                                     float xx, float xy, float rx, float ry,
                                     const Consts& K) {
  // adj = h*(1+r) / (1 + hx*rx + hy*ry)
  float ux = fmaf(hx, rx, hx);                   // hx*(1+rx), parallel with s/rcp
  float uy = fmaf(hy, ry, hy);
  float s = fmaf(hy, ry, fmaf(hx, rx, 1.0f));
  float inv = __builtin_amdgcn_rcpf(s);          // v_rcp_f32 (TRANS, co-executes)
  inv = inv * fmaf(-s, inv, 2.0f);               // one Newton step
  float ax = ux * inv;
  float ay = uy * inv;
  // corner-clamped bounds; (a - c)*slope pre-folded to fma(a, slope, -c*slope)
  float lbx = K.base_lbx - fmaxf(K.A_lbx - fmaxf(fmaf(ay, K.bd, K.ncs_lbx), 0.0f), 0.0f);
  float ubx = K.base_ubx - fmaxf(K.A_ubx - fmaxf(fmaf(ay, K.bd, K.ncs_ubx), 0.0f), 0.0f);
  float lby = K.base_lby - fmaxf(K.A_lby - fmaxf(fmaf(ax, K.ac, K.ncs_lby), 0.0f), 0.0f);
  float uby = K.base_uby - fmaxf(K.A_uby - fmaxf(fmaf(ax, K.ac, K.ncs_uby), 0.0f), 0.0f);
  // g1 = relu(Wi*x + Wh*adj - lb); g2 = relu(W1*g1 + ub - lb); h = W2*g2 + ub
  float g1x = fmaxf(fmaf(K.Wh0, ax, fmaf(K.Wi0, xx, -lbx)), 0.0f);
  float g1y = fmaxf(fmaf(K.Wh1, ay, fmaf(K.Wi1, xy, -lby)), 0.0f);
  float g2x = fmaxf(fmaf(K.W10, g1x, ubx - lbx), 0.0f);
  float g2y = fmaxf(fmaf(K.W11, g1y, uby - lby), 0.0f);
  hx = fmaf(K.W20, g2x, ubx);
  hy = fmaf(K.W21, g2y, uby);
}

__global__ __launch_bounds__(32) void notrade_scan_kernel(
    const float* __restrict__ x, const float* __restrict__ r,
    const float* __restrict__ tgt, const float* __restrict__ wi,
    const float* __restrict__ wh, const float* __restrict__ bh,
    const float* __restrict__ w1, const float* __restrict__ w2,
    const float* __restrict__ wr, float* __restrict__ out,
    int T, int B) {
  const int lane = threadIdx.x;
  const int b = blockIdx.x * 32 + lane;
  if (b >= B) return;

  // ---- uniform weights -> scalar loads; fold corner logic into constants ----
  const float tx = tgt[0], ty = tgt[1];
  const float b0 = bh[0], b1 = bh[1];
  const float R00 = wr[0], R01 = wr[1], R10 = wr[2], R11 = wr[3];
  Consts K;
  K.ac = R10 / R00;
  K.bd = R01 / R11;
  const float p0 = R00 * b0, p1 = R01 * b1, q0 = R10 * b0, q1 = R11 * b1;
  const float C0x = -p0 - p1 + tx, C0y = -q0 - q1 + ty;
  const float C1x = -p0 + p1 + tx, C1y = -q0 + q1 + ty;
  const float C2x =  p0 + p1 + tx, C2y =  q0 + q1 + ty;
  const float C3x =  p0 - p1 + tx, C3y =  q0 - q1 + ty;
  const bool bdp = (K.bd >= 0.0f), acp = (K.ac >= 0.0f);
  K.ncs_lbx = -(bdp ? C0y : C1y) * K.bd;  K.A_lbx = fabsf(C1x - C0x);  K.base_lbx = bdp ? C1x : C0x;
  K.ncs_ubx = -(bdp ? C3y : C2y) * K.bd;  K.A_ubx = fabsf(C2x - C3x);  K.base_ubx = bdp ? C2x : C3x;
  K.ncs_lby = -(acp ? C0x : C3x) * K.ac;  K.A_lby = fabsf(C0y - C3y);  K.base_lby = acp ? C3y : C0y;
  K.ncs_uby = -(acp ? C1x : C2x) * K.ac;  K.A_uby = fabsf(C1y - C2y);  K.base_uby = acp ? C2y : C1y;
  K.Wi0 = wi[0]; K.Wi1 = wi[1]; K.Wh0 = wh[0]; K.Wh1 = wh[1];
  K.W10 = w1[0]; K.W11 = w1[1]; K.W20 = w2[0]; K.W21 = w2[1];

  const size_t TB = (size_t)T * (size_t)B;
  const float* x0 = x;        const float* x1 = x + TB;   // (D,T,B): d-planes TB apart
  const float* r0 = r;        const float* r1 = r + TB;
  float* o0 = out;            float* o1 = out + TB;
  float* hT = out + 2 * TB;

  // h0 = x[:,0,:]; output[:,0,:] = h0
  float hx = x0[b], hy = x1[b];
  __builtin_nontemporal_store(hx, &o0[b]);
  __builtin_nontemporal_store(hy, &o1[b]);

#if USE_ASYNC
  // -------- async global->LDS 8-stage pipeline (4 KB per single-wave WG) --------
  __shared__ float sbuf[STAGES][4][32];
  auto issue = [&](int t, int slot) {
    const size_t ox = (size_t)t * B + b;         // x at step t
    const size_t orr = (size_t)(t - 1) * B + b;  // returns at step t-1
    async_ld_b32(&x0[ox], &sbuf[slot][0][lane]);
    async_ld_b32(&x1[ox], &sbuf[slot][1][lane]);
    async_ld_b32(&r0[orr], &sbuf[slot][2][lane]);
    async_ld_b32(&r1[orr], &sbuf[slot][3][lane]);
  };
  // prologue: fill all stages (t = 1..STAGES), slot = (t-1)&(STAGES-1)
  for (int i = 0; i < STAGES; ++i) {
    int t = 1 + i;
    if (t > T - 1) t = T - 1;
    issue(t, i & (STAGES - 1));
  }
  for (int t = 1; t < T; ++t) {
    const int slot = (t - 1) & (STAGES - 1);
    wait_async_oldest_stage();                        // oldest stage landed in LDS
    // force real ds_load_b32 via explicit LDS-addrspace volatile reads
    float xx = *(lds_vcf*)&sbuf[slot][0][lane];
    float xy = *(lds_vcf*)&sbuf[slot][1][lane];
    float rx = *(lds_vcf*)&sbuf[slot][2][lane];
    float ry = *(lds_vcf*)&sbuf[slot][3][lane];
    wait_ds_zero();                                   // LDS reads done before slot reuse
    int tf = t + STAGES;                              // keep ASYNCcnt constant in tail:
    if (tf > T - 1) tf = T - 1;                       // clamped in-bounds dummy issues
    issue(tf, slot);
    step(hx, hy, xx, xy, rx, ry, K);
    const size_t oo = (size_t)t * B + b;
    __builtin_nontemporal_store(hx, &o0[oo]);
    __builtin_nontemporal_store(hy, &o1[oo]);
  }
#else
  // -------- fallback: direct loads + global_prefetch_b8 look-ahead --------
  const int PF = 12;
  for (int t = 1; t < T; ++t) {
    const size_t oo = (size_t)t * B + b;
    const size_t orr = oo - (size_t)B;
    if (t + PF < T) {
      const size_t po = (size_t)(t + PF) * B + b;
      __builtin_prefetch(&x0[po], 0, 0);
      __builtin_prefetch(&x1[po], 0, 0);
      __builtin_prefetch(&r0[po - (size_t)B], 0, 0);
      __builtin_prefetch(&r1[po - (size_t)B], 0, 0);
    }
    float xx = x0[oo], xy = x1[oo];
    float rx = r0[orr], ry = r1[orr];
    step(hx, hy, xx, xy, rx, ry, K);
    __builtin_nontemporal_store(hx, &o0[oo]);
    __builtin_nontemporal_store(hy, &o1[oo]);
  }
#endif

  // hT = final carry, appended after the (D,T,B) output block
  hT[b] = hx;
  hT[B + b] = hy;
}

extern "C" void kernel_launch(void* const* d_in, const int* in_sizes, int n_in,
                              void* d_out, int out_size, void* d_ws, size_t ws_size,
                              hipStream_t stream) {
  // dict order: input, target, returns_partition, hidden(unused), w_input,
  //             w_hidden, b_hidden, w_fc1, w_fc2, w_rotate
  const float* x  = (const float*)d_in[0];
  const float* tg = (const float*)d_in[1];
  const float* r  = (const float*)d_in[2];
  const float* wi = (const float*)d_in[4];
  const float* wh = (const float*)d_in[5];
  const float* bh = (const float*)d_in[6];
  const float* w1 = (const float*)d_in[7];
  const float* w2 = (const float*)d_in[8];
  const float* wr = (const float*)d_in[9];

  const int B = in_sizes[3] / 2;            // hidden is (D,1,B) = 2*B
  const int T = in_sizes[0] / (2 * B);      // input is (D,T,B)

  dim3 block(32);
  dim3 grid((B + 31) / 32);                 // 512 single-wave workgroups
  notrade_scan_kernel<<<grid, block, 0, stream>>>(
      x, r, tg, wi, wh, bh, w1, w2, wr, (float*)d_out, T, B);
}